// SAGE_16329465660094
// MI455X (gfx1250) — compile-verified
//
#include <hip/hip_runtime.h>
#include <hip/hip_bf16.h>

#define N_NODES 50000
#define N_EDGES 800000
#define IN_F 96
#define HID_F 96
#define OUT_F 32

typedef __attribute__((ext_vector_type(2))) float v2f;
typedef __attribute__((ext_vector_type(8))) float v8f;

// ---------------------------------------------------------------------------
// Degree: one f32 atomic per edge.
// ---------------------------------------------------------------------------
__global__ void sage_deg_kernel(const int* __restrict__ dst,
                                float* __restrict__ deg, int n_edges) {
    int e = blockIdx.x * blockDim.x + threadIdx.x;
    if (e >= n_edges) return;
    atomicAdd(&deg[dst[e]], 1.0f);
}

// ---------------------------------------------------------------------------
// Scatter-add of 96-wide feature rows: one thread per (edge, float4 chunk).
// 24 chunks of float4 cover 96 floats. Gathers are 16B coalesced; writes are
// GLOBAL_ATOMIC_ADD_F32.
// ---------------------------------------------------------------------------
__global__ void sage_scatter_kernel(const float* __restrict__ x,
                                    const int* __restrict__ src,
                                    const int* __restrict__ dst,
                                    float* __restrict__ msg, int n_edges) {
    int tid = blockIdx.x * blockDim.x + threadIdx.x;
    int e = tid / 24;
    int c = tid % 24;
    if (e >= n_edges) return;
    int s = src[e];
    int d = dst[e];
    float4 v = ((const float4*)(x + (size_t)s * 96))[c];
    float* m = msg + (size_t)d * 96 + (size_t)c * 4;
    atomicAdd(m + 0, v.x);
    atomicAdd(m + 1, v.y);
    atomicAdd(m + 2, v.z);
    atomicAdd(m + 3, v.w);
}

// ---------------------------------------------------------------------------
// Fused SAGE layer GEMM on WMMA f32 16x16x4:
//   out = act( A*Ws + (An .* invdeg_row)*Wn + bias )
// One wave per 16x16 output tile. K and the deg-normalization are fused into
// the A-operand register build, eliminating a separate normalize pass.
//
// A-operand layout (ISA 16x4 f32): lanes 0-15 carry rows M=0..15 with
// {K=k, k+1} in {v0,v1}; lanes 16-31 carry the same rows with {K=k+2, k+3}.
// B-operand mirrors: v0 = B[k+2*half][n], v1 = B[k+1+2*half][n].
// C/D layout: VGPR i -> row (i + 8*half), col = lane&15.
// ---------------------------------------------------------------------------
__global__ void sage_gemm_wmma(const float* __restrict__ A,    // [M,K] self
                               const float* __restrict__ An,   // [M,K] neighbor sums
                               const float* __restrict__ deg,  // [M]
                               const float* __restrict__ Ws,   // [K,N]
                               const float* __restrict__ Wn,   // [K,N]
                               const float* __restrict__ bias, // [N]
                               float* __restrict__ out,        // [M,N]
                               int K, int N, int nTilesN, int nTiles,
                               int do_relu) {
    int wave = blockIdx.x * (blockDim.x >> 5) + (threadIdx.x >> 5);
    if (wave >= nTiles) return;  // wave-uniform: EXEC stays all-ones for WMMA

    int lane = threadIdx.x & 31;
    int half = lane >> 4;   // 0: K=k,k+1   1: K=k+2,k+3
    int l    = lane & 15;

    int mt = wave / nTilesN;
    int nt = wave % nTilesN;
    int row  = mt * 16 + l;
    int ncol = nt * 16 + l;

    const float* arow = A  + (size_t)row * K;
    const float* nrow = An + (size_t)row * K;
    float invd = 1.0f / fmaxf(deg[row], 1.0f);

    v8f c = {};
    // self-path: A * Ws
    for (int k = 0; k < K; k += 4) {
        int ko = k + 2 * half;
        v2f a = {arow[ko], arow[ko + 1]};
        v2f b = {Ws[(size_t)ko * N + ncol], Ws[(size_t)(ko + 1) * N + ncol]};
        c = __builtin_amdgcn_wmma_f32_16x16x4_f32(false, a, false, b,
                                                  (short)0, c, false, false);
    }
    // neighbor-path: (An * invdeg) * Wn, normalization fused into operand build
    for (int k = 0; k < K; k += 4) {
        int ko = k + 2 * half;
        v2f a = {nrow[ko] * invd, nrow[ko + 1] * invd};
        v2f b = {Wn[(size_t)ko * N + ncol], Wn[(size_t)(ko + 1) * N + ncol]};
        c = __builtin_amdgcn_wmma_f32_16x16x4_f32(false, a, false, b,
                                                  (short)0, c, false, false);
    }

    float bn = bias[ncol];
#pragma unroll
    for (int i = 0; i < 8; i++) {
        float v = c[i] + bn;
        if (do_relu) v = fmaxf(v, 0.0f);
        out[(size_t)(mt * 16 + i + 8 * half) * N + ncol] = v;
    }
}

// ---------------------------------------------------------------------------
// Orchestration. Workspace layout (floats):
//   deg : [0, 50176)                      (padded for alignment)
//   msg : [50176, 50176 + 4'800'000)
//   h   : [.. + 4'800'000)
// Total ~38.6 MB. All zeroing is done per call (deterministic).
// ---------------------------------------------------------------------------
extern "C" void kernel_launch(void* const* d_in, const int* in_sizes, int n_in,
                              void* d_out, int out_size, void* d_ws,
                              size_t ws_size, hipStream_t stream) {
    const float* x        = (const float*)d_in[0];
    const float* w_self1  = (const float*)d_in[1];
    const float* w_neigh1 = (const float*)d_in[2];
    const float* b1       = (const float*)d_in[3];
    const float* w_self2  = (const float*)d_in[4];
    const float* w_neigh2 = (const float*)d_in[5];
    const float* b2       = (const float*)d_in[6];
    const int*   src      = (const int*)d_in[7];
    const int*   dst      = (const int*)d_in[8];
    float*       out      = (float*)d_out;

    float* deg = (float*)d_ws;
    float* msg = deg + 50176;
    float* h   = msg + (size_t)N_NODES * HID_F;

    const size_t msg_bytes = (size_t)N_NODES * HID_F * sizeof(float);

    // --- layer 1 aggregation ---
    hipMemsetAsync(deg, 0, 50176 * sizeof(float), stream);
    hipMemsetAsync(msg, 0, msg_bytes, stream);
    sage_deg_kernel<<<(N_EDGES + 255) / 256, 256, 0, stream>>>(dst, deg, N_EDGES);
    {
        long long work = (long long)N_EDGES * 24;
        sage_scatter_kernel<<<(int)((work + 255) / 256), 256, 0, stream>>>(
            x, src, dst, msg, N_EDGES);
    }
    // --- layer 1 fused GEMM + bias + relu ---
    {
        int nTilesN = HID_F / 16;                    // 6
        int nTiles  = (N_NODES / 16) * nTilesN;      // 18750
        sage_gemm_wmma<<<(nTiles + 7) / 8, 256, 0, stream>>>(
            x, msg, deg, w_self1, w_neigh1, b1, h, IN_F, HID_F, nTilesN,
            nTiles, 1);
    }

    // --- layer 2 aggregation ---
    hipMemsetAsync(msg, 0, msg_bytes, stream);
    {
        long long work = (long long)N_EDGES * 24;
        sage_scatter_kernel<<<(int)((work + 255) / 256), 256, 0, stream>>>(
            h, src, dst, msg, N_EDGES);
    }
    // --- layer 2 fused GEMM + bias (no relu) ---
    {
        int nTilesN = OUT_F / 16;                    // 2
        int nTiles  = (N_NODES / 16) * nTilesN;      // 6250
        sage_gemm_wmma<<<(nTiles + 7) / 8, 256, 0, stream>>>(
            h, msg, deg, w_self2, w_neigh2, b2, out, HID_F, OUT_F, nTilesN,
            nTiles, 0);
    }
}